// RealAgnosticInteractionBlock_33303176413872
// MI455X (gfx1250) — compile-verified
//
#include <hip/hip_runtime.h>
#include <hip/hip_bf16.h>

typedef __attribute__((ext_vector_type(2))) float v2f;
typedef __attribute__((ext_vector_type(4))) float f4;
typedef __attribute__((ext_vector_type(8))) float v8f;
typedef __attribute__((ext_vector_type(4))) unsigned int v4u;
typedef __attribute__((ext_vector_type(4))) int v4i;
typedef __attribute__((ext_vector_type(8))) int v8i;

#if __has_builtin(__builtin_amdgcn_tensor_load_to_lds) && __has_builtin(__builtin_amdgcn_s_wait_tensorcnt)
#define HAVE_TDM 1
#else
#define HAVE_TDM 0
#endif

// fp32 WMMA: D = A(16x4) * B(4x16) + C(16x16), full fp32 precision.
__device__ __forceinline__ v8f wmma_f32(v2f a, v2f b, v8f c) {
    return __builtin_amdgcn_wmma_f32_16x16x4_f32(
        false, a, false, b, (short)0, c, false, false);
}

__device__ __forceinline__ float silu(float x) {
    return x / (1.0f + __expf(-x));
}

#if HAVE_TDM
// TDM gather-mode: pull 16 rows (128 f32 each) selected by 16-bit row indices
// from a row-major (rows_total x 128) tensor into LDS, asynchronously.
// 6-arg builtin form (amdgpu-toolchain / clang-23).
__device__ __forceinline__ void tdm_gather16_rows(const float* gbase,
                                                  unsigned lds_byte,
                                                  v4i idxlo, v4i idxhi,
                                                  int rows_total)
{
    const unsigned long long ga = (unsigned long long)(uintptr_t)gbase;
    v4u g0;
    g0[0] = 0x80000001u;                         // gather_mode=1, 16b idx, count=1
    g0[1] = lds_byte;                            // lds_addr
    g0[2] = (unsigned)(ga & 0xFFFFFFFFu);        // global_addr[31:0]
    g0[3] = (unsigned)((ga >> 32) & 0x1FFFFFFu)  // global_addr[56:32]
          | (2u << 30);                          // type = 2 (image)
    v8i g1;
    g1[0] = (int)(2u << 16);                     // wg_mask=0, data_size=4B
    g1[1] = (int)(128u << 16);                   // tensor_dim0 = 128
    g1[2] = (int)((unsigned)rows_total << 16);   // tensor_dim1[15:0]
    g1[3] = (int)(128u << 16)                    // tile_dim0 = 128
          | (int)((unsigned)rows_total >> 16);   // tensor_dim1[31:16]
    g1[4] = 16;                                  // tile_dim1 = 16 gather indices
    g1[5] = 128;                                 // tensor_dim0_stride = 128
    g1[6] = 0;
    g1[7] = 0;
    v8i gx = {};                                 // unused trailing operand group
    __builtin_amdgcn_tensor_load_to_lds(g0, g1, idxlo, idxhi, gx, 0);
}
#endif

// ---------------------------------------------------------------------------
// Generic tiled GEMM: C[M,N] = alpha * A @ B
//   A element (m,k) at A[m*lda + k*astride]; B row-major (K,N).
//   One wave per 16x16 C tile.
// ---------------------------------------------------------------------------
__global__ void wmma_gemm_f32(const float* __restrict__ A, int lda, int astride,
                              const float* __restrict__ B, int ldb,
                              float* __restrict__ C, int ldc,
                              int K, float alpha)
{
    const int n0 = blockIdx.x * 16;
    const int m0 = blockIdx.y * 16;
    const int lane = threadIdx.x;
    const int half = lane >> 4;      // fp32 frag layout: half selects K+0/1 vs K+2/3
    const int lr   = lane & 15;

    const float* Arow = A + (size_t)(m0 + lr) * lda;
    const float* Bcol = B + (n0 + lr);

    v8f c = {};
    for (int k0 = 0; k0 < K; k0 += 4) {
        const int ka = k0 + (half ? 2 : 0);
        v2f a, b;
        a.x = Arow[(size_t)ka * astride];
        a.y = Arow[(size_t)(ka + 1) * astride];
        b.x = Bcol[(size_t)ka * ldb];
        b.y = Bcol[(size_t)(ka + 1) * ldb];
        c = wmma_f32(a, b, c);
    }
#pragma unroll
    for (int r = 0; r < 8; ++r) {
        const int m = m0 + r + (half ? 8 : 0);
        C[(size_t)m * ldc + n0 + lr] = alpha * c[r];
    }
}

// ---------------------------------------------------------------------------
// Edge MLP stages 1+2: h2 = silu( silu(ef @ W1 /sqrt8) @ W2 /sqrt64 )
// One wave per 16 edges. MLP1 (K=8) by VALU into LDS, MLP2 (K=64) by WMMA.
// ---------------------------------------------------------------------------
__global__ void edge_mlp_kernel(const float* __restrict__ ef,   // E x 8
                                const float* __restrict__ W1,   // 8 x 64
                                const float* __restrict__ W2,   // 64 x 64
                                float* __restrict__ h2out)      // E x 64
{
    __shared__ float h1[16 * 64];
    const int e0 = blockIdx.x * 16;
    const int lane = threadIdx.x;

    const float inv1 = 0.3535533905932738f;  // 1/sqrt(8)
    for (int idx = lane; idx < 16 * 64; idx += 32) {
        const int e = idx >> 6, j = idx & 63;
        const float* efr = ef + (size_t)(e0 + e) * 8;
        float acc = 0.0f;
#pragma unroll
        for (int k = 0; k < 8; ++k) acc += efr[k] * W1[k * 64 + j];
        h1[idx] = silu(acc * inv1);
    }
    __syncthreads();

    const int half = lane >> 4, lr = lane & 15;
    const float inv2 = 0.125f;               // 1/sqrt(64)
    for (int nt = 0; nt < 4; ++nt) {
        const int n0 = nt * 16;
        v8f c = {};
        for (int k0 = 0; k0 < 64; k0 += 4) {
            const int ka = k0 + (half ? 2 : 0);
            v2f a, b;
            a.x = h1[lr * 64 + ka];
            a.y = h1[lr * 64 + ka + 1];
            b.x = W2[ka * 64 + n0 + lr];
            b.y = W2[(ka + 1) * 64 + n0 + lr];
            c = wmma_f32(a, b, c);
        }
#pragma unroll
        for (int r = 0; r < 8; ++r) {
            const int m = r + (half ? 8 : 0);
            h2out[(size_t)(e0 + m) * 64 + n0 + lr] = silu(c[r] * inv2);
        }
    }
}

// ---------------------------------------------------------------------------
// Fused MLP3 + tensor-product messages + scatter-add.
// Block = 128 threads (4 waves) per 16 edges. Gathered sender features arrive
// via TDM gather-mode DMA (TENSORcnt) when available; WMMA computes the tpw
// chunk per wave into LDS; messages are atomically accumulated into Ms/Mv.
// ---------------------------------------------------------------------------
__global__ void edge_msg_kernel(const float* __restrict__ h2,   // E x 64
                                const float* __restrict__ W3,   // 64 x 512
                                const float* __restrict__ edge_attrs, // E x 4
                                const int*   __restrict__ edge_index, // 2 x E
                                const float* __restrict__ s_up, // N x 128
                                const float* __restrict__ v_up, // 3 x N x 128
                                float* __restrict__ Ms,         // N x 256
                                float* __restrict__ Mv,         // 3 x N x 256
                                int E, int Nn)
{
    __shared__ float h2s[16 * 64];
    __shared__ float xs[16 * 128];
    __shared__ float xv[3][16 * 128];
    __shared__ float ea[16 * 4];
    __shared__ int   recv[16];
    __shared__ float tpw[4][16 * 128];

    const int e0 = blockIdx.x * 16;
    const int t = threadIdx.x;               // 0..127
    const size_t NPLANE = (size_t)Nn * 128;
    const size_t MPLANE = (size_t)Nn * 256;

    // Warm the W_mlp3 stripe this block streams (global_prefetch_b8).
    __builtin_prefetch(W3 + (size_t)(t & 63) * 512 + (t >> 6) * 256, 0, 0);

    if (t < 16) recv[t] = edge_index[E + e0 + t];
    for (int idx = t; idx < 16 * 4; idx += 128)
        ea[idx] = edge_attrs[(size_t)e0 * 4 + idx];
    for (int idx = t; idx < 16 * 16; idx += 128)        // h2 tile: 256 x float4
        ((f4*)h2s)[idx] = ((const f4*)(h2 + (size_t)e0 * 64))[idx];

#if HAVE_TDM
    if (t < 32) {
        int snd[16];
#pragma unroll
        for (int e = 0; e < 16; ++e) snd[e] = edge_index[e0 + e];
        v4i ilo, ihi;
#pragma unroll
        for (int q = 0; q < 4; ++q) {
            ilo[q] = (snd[2 * q]     & 0xFFFF) | (snd[2 * q + 1] << 16);
            ihi[q] = (snd[8 + 2 * q] & 0xFFFF) | (snd[9 + 2 * q] << 16);
        }
        tdm_gather16_rows(s_up,              (unsigned)(uintptr_t)&xs[0],    ilo, ihi, Nn);
        tdm_gather16_rows(v_up,              (unsigned)(uintptr_t)&xv[0][0], ilo, ihi, Nn);
        tdm_gather16_rows(v_up + NPLANE,     (unsigned)(uintptr_t)&xv[1][0], ilo, ihi, Nn);
        tdm_gather16_rows(v_up + 2 * NPLANE, (unsigned)(uintptr_t)&xv[2][0], ilo, ihi, Nn);
        asm volatile("" ::: "memory");       // LDS is written by the DMA engine
        __builtin_amdgcn_s_wait_tensorcnt(0);
    }
#else
    for (int idx = t; idx < 16 * 32; idx += 128) {      // 16 rows x 32 float4
        const int e = idx >> 5, q = idx & 31;
        const int snd = edge_index[e0 + e];
        const size_t base = (size_t)snd * 128;
        ((f4*)xs)[e * 32 + q]    = ((const f4*)(s_up + base))[q];
        ((f4*)xv[0])[e * 32 + q] = ((const f4*)(v_up + base))[q];
        ((f4*)xv[1])[e * 32 + q] = ((const f4*)(v_up + NPLANE + base))[q];
        ((f4*)xv[2])[e * 32 + q] = ((const f4*)(v_up + 2 * NPLANE + base))[q];
    }
#endif
    __syncthreads();

    const int wave = t >> 5;
    const int lane = t & 31;
    const int half = lane >> 4, lr = lane & 15;
    float* tp = tpw[wave];
    const float inv3 = 0.125f;               // 1/sqrt(64)

    // tpw chunk `wave` (16 edges x 128 cols) = h2s(16x64) @ W3[:, wave*128 ...]
    for (int nt = 0; nt < 8; ++nt) {
        const int n0 = nt * 16;
        const float* Bc = W3 + (size_t)wave * 128 + n0 + lr;
        v8f c = {};
        for (int k0 = 0; k0 < 64; k0 += 4) {
            const int ka = k0 + (half ? 2 : 0);
            v2f a, b;
            a.x = h2s[lr * 64 + ka];
            a.y = h2s[lr * 64 + ka + 1];
            b.x = Bc[(size_t)ka * 512];
            b.y = Bc[(size_t)(ka + 1) * 512];
            c = wmma_f32(a, b, c);
        }
#pragma unroll
        for (int r = 0; r < 8; ++r)
            tp[(r + (half ? 8 : 0)) * 128 + n0 + lr] = c[r] * inv3;
    }
    __syncthreads();

    const float SQ3INV = 0.5773502691896258f; // 1/sqrt(3)
    for (int idx = lane; idx < 16 * 128; idx += 32) {
        const int e = idx >> 7, u = idx & 127;
        const int rn = recv[e];
        const float w = tp[idx];
        const float eas = ea[e * 4];
        const float ev0 = ea[e * 4 + 1], ev1 = ea[e * 4 + 2], ev2 = ea[e * 4 + 3];
        if (wave == 0) {            // m0 = w_a * xs * ea_s  -> Ms[:, :128]
            unsafeAtomicAdd(&Ms[(size_t)rn * 256 + u], w * xs[idx] * eas);
        } else if (wave == 1) {     // m1 = w_b * (xv . ea_v)/sqrt3 -> Ms[:,128:]
            const float dot = xv[0][idx] * ev0 + xv[1][idx] * ev1 + xv[2][idx] * ev2;
            unsafeAtomicAdd(&Ms[(size_t)rn * 256 + 128 + u], w * dot * SQ3INV);
        } else if (wave == 2) {     // m2 = (w_c*xs) outer ea_v -> Mv[:, :128, i]
            const float txs = w * xs[idx];
            unsafeAtomicAdd(&Mv[(size_t)rn * 256 + u],              txs * ev0);
            unsafeAtomicAdd(&Mv[MPLANE + (size_t)rn * 256 + u],     txs * ev1);
            unsafeAtomicAdd(&Mv[2 * MPLANE + (size_t)rn * 256 + u], txs * ev2);
        } else {                    // m3 = w_d * xv * ea_s -> Mv[:, 128:, i]
            unsafeAtomicAdd(&Mv[(size_t)rn * 256 + 128 + u],              w * xv[0][idx] * eas);
            unsafeAtomicAdd(&Mv[MPLANE + (size_t)rn * 256 + 128 + u],     w * xv[1][idx] * eas);
            unsafeAtomicAdd(&Mv[2 * MPLANE + (size_t)rn * 256 + 128 + u], w * xv[2][idx] * eas);
        }
    }
}

// ---------------------------------------------------------------------------
// Species-conditioned skip: C[n,w] = alpha * sum_v sum_u A[n,u]*attrs[n,v]*Wsk[u,v,w]
// Row-scale folded into the A-fragment load; accumulate over 4 species.
// ---------------------------------------------------------------------------
__global__ void skip_wmma_kernel(const float* __restrict__ A,
                                 const float* __restrict__ attrs,  // N x 4
                                 const float* __restrict__ Wsk,    // (128,4,128)
                                 float* __restrict__ C,
                                 float alpha)
{
    const int n0 = blockIdx.x * 16;
    const int m0 = blockIdx.y * 16;
    const int lane = threadIdx.x;
    const int half = lane >> 4, lr = lane & 15;
    const int m = m0 + lr;
    const float* Arow = A + (size_t)m * 128;

    v8f c = {};
#pragma unroll 1
    for (int sp = 0; sp < 4; ++sp) {
        const float rs = attrs[m * 4 + sp];
        const float* Bc = Wsk + sp * 128 + n0 + lr;
        for (int k0 = 0; k0 < 128; k0 += 4) {
            const int ka = k0 + (half ? 2 : 0);
            v2f a, b;
            a.x = Arow[ka] * rs;
            a.y = Arow[ka + 1] * rs;
            b.x = Bc[(size_t)ka * 512];
            b.y = Bc[(size_t)(ka + 1) * 512];
            c = wmma_f32(a, b, c);
        }
    }
#pragma unroll
    for (int r = 0; r < 8; ++r)
        C[(size_t)(m0 + r + (half ? 8 : 0)) * 128 + n0 + lr] = alpha * c[r];
}

// ---------------------------------------------------------------------------
__global__ void zero_kernel4(f4* __restrict__ p, size_t n4) {
    const size_t i = (size_t)blockIdx.x * blockDim.x + threadIdx.x;
    if (i < n4) p[i] = (f4){0.0f, 0.0f, 0.0f, 0.0f};
}

// out[n,u,:] = {c_old*S + c_new*scs, c_old*V_i + c_new*scv_i} as one b128 store
__global__ void combine_kernel(const float* __restrict__ S,
                               const float* __restrict__ scs,
                               const float* __restrict__ V,    // 3 planes N*128
                               const float* __restrict__ scv,  // 3 planes N*128
                               const float* __restrict__ param,
                               float* __restrict__ out, int Nn)
{
    const int idx = blockIdx.x * blockDim.x + threadIdx.x;
    const int total = Nn * 128;
    if (idx >= total) return;
    const float cc = 1.0f / (1.0f + __expf(-param[0]));
    const float c_old = rsqrtf(cc * cc + 1.0f);
    const float c_new = cc * c_old;
    const size_t NP = (size_t)Nn * 128;
    f4 o;
    o.x = c_old * S[idx] + c_new * scs[idx];
    o.y = c_old * V[idx] + c_new * scv[idx];
    o.z = c_old * V[NP + idx] + c_new * scv[NP + idx];
    o.w = c_old * V[2 * NP + idx] + c_new * scv[2 * NP + idx];
    ((f4*)out)[idx] = o;
}

// ---------------------------------------------------------------------------
extern "C" void kernel_launch(void* const* d_in, const int* in_sizes, int n_in,
                              void* d_out, int out_size, void* d_ws, size_t ws_size,
                              hipStream_t stream)
{
    const float* node_attrs = (const float*)d_in[0];
    const float* node_feats = (const float*)d_in[1];   // N x 512
    const float* edge_attrs = (const float*)d_in[2];   // E x 4
    const float* edge_feats = (const float*)d_in[3];   // E x 8
    const int*   edge_index = (const int*)d_in[4];     // 2 x E
    const float* W_up0  = (const float*)d_in[5];
    const float* W_up1  = (const float*)d_in[6];
    const float* W_mlp1 = (const float*)d_in[7];
    const float* W_mlp2 = (const float*)d_in[8];
    const float* W_mlp3 = (const float*)d_in[9];
    const float* W_lin0 = (const float*)d_in[10];
    const float* W_lin1 = (const float*)d_in[11];
    const float* W_skip0 = (const float*)d_in[12];
    const float* W_skip1 = (const float*)d_in[13];
    const float* param   = (const float*)d_in[14];

    const int Nn = 10000, E = 160000;
    const size_t NP = (size_t)Nn * 128;

    float* ws   = (float*)d_ws;
    float* s_up = ws;                 // NP
    float* v_up = ws + NP;            // 3*NP
    float* Ms   = ws + 4 * NP;        // 2*NP
    float* Mv   = ws + 6 * NP;        // 6*NP
    float* h2   = ws + 12 * NP;       // 8*NP == E*64
    // After h2 is consumed, reuse its region:
    float* S    = ws + 12 * NP;       // NP
    float* V    = ws + 13 * NP;       // 3*NP
    float* scs  = ws + 16 * NP;       // NP
    float* scv  = ws + 17 * NP;       // 3*NP  (total ws use: 20*NP floats)

    const float inv_up  = 0.08838834764831845f;  // 1/sqrt(128)
    const float inv_lin = 0.00390625f;           // 1/(sqrt(256)*16)
    const float inv_sk  = 0.04419417382415922f;  // 1/sqrt(512)

    dim3 g128(128 / 16, Nn / 16);

    // Node up-projections (fp32 WMMA GEMMs)
    wmma_gemm_f32<<<g128, 32, 0, stream>>>(node_feats, 512, 1, W_up0, 128,
                                           s_up, 128, 128, inv_up);
    for (int i = 0; i < 3; ++i)
        wmma_gemm_f32<<<g128, 32, 0, stream>>>(node_feats + 128 + i, 512, 3,
                                               W_up1, 128, v_up + i * NP, 128,
                                               128, inv_up);

    // Edge MLP stages 1+2
    edge_mlp_kernel<<<E / 16, 32, 0, stream>>>(edge_feats, W_mlp1, W_mlp2, h2);

    // Zero segment-sum accumulators (Ms ++ Mv contiguous: 8*NP floats)
    {
        const size_t n4 = 8 * NP / 4;
        zero_kernel4<<<(unsigned)((n4 + 255) / 256), 256, 0, stream>>>((f4*)Ms, n4);
    }

    // Fused MLP3 + messages + scatter (TDM gather of sender features)
    edge_msg_kernel<<<E / 16, 128, 0, stream>>>(h2, W_mlp3, edge_attrs, edge_index,
                                                s_up, v_up, Ms, Mv, E, Nn);

    // Linear layers on aggregated messages
    wmma_gemm_f32<<<g128, 32, 0, stream>>>(Ms, 256, 1, W_lin0, 128,
                                           S, 128, 256, inv_lin);
    for (int i = 0; i < 3; ++i)
        wmma_gemm_f32<<<g128, 32, 0, stream>>>(Mv + i * 2 * NP, 256, 1, W_lin1, 128,
                                               V + i * NP, 128, 256, inv_lin);

    // Species-conditioned skip connections
    skip_wmma_kernel<<<g128, 32, 0, stream>>>(S, node_attrs, W_skip0, scs, inv_sk);
    for (int i = 0; i < 3; ++i)
        skip_wmma_kernel<<<g128, 32, 0, stream>>>(V + i * NP, node_attrs, W_skip1,
                                                  scv + i * NP, inv_sk);

    // Final sigmoid-weighted combine + interleave into (N, 128, 4)
    combine_kernel<<<(Nn * 128 + 255) / 256, 256, 0, stream>>>(
        S, scs, V, scv, param, (float*)d_out, Nn);
}